// InterationAttention_3332894622708
// MI455X (gfx1250) — compile-verified
//
#include <hip/hip_runtime.h>
#include <hip/hip_bf16.h>
#include <math.h>

// Problem constants (match reference)
constexpr int kB  = 8;
constexpr int kN  = 128;
constexpr int kM  = 256;
constexpr int kF1 = 256;
constexpr int kF2 = 256;

typedef __attribute__((ext_vector_type(2))) float v2f;
typedef __attribute__((ext_vector_type(8))) float v8f;

// ---------------------------------------------------------------------------
// Hardware tanh: CDNA5 has a TRANS32 TANH op. Prefer the builtin so we emit
// v_tanh_f32 instead of a libm polynomial (67M tanh evals dominate runtime).
// ---------------------------------------------------------------------------
__device__ __forceinline__ float fast_tanh(float v) {
#if __has_builtin(__builtin_amdgcn_tanhf)
  return __builtin_amdgcn_tanhf(v);
#elif __has_builtin(__builtin_amdgcn_tanh_f32)
  return __builtin_amdgcn_tanh_f32(v);
#else
  return tanhf(v);
#endif
}

// ---------------------------------------------------------------------------
// Kernel 1: u[bn, g] = sum_f x[bn, f] * Uw[g, f] + Ub[g]
// One wave32 per 16x16 tile of u. V_WMMA_F32_16X16X4_F32, K-loop of 64.
//
// VGPR layouts (ISA 7.12.2, f32):
//   A (16x4, MxK): lane L row = L%16; VGPR0 holds K = 2*(L/16), VGPR1 K = +1.
//   B (4x16, KxN): mirrored — lane L col = L%16; VGPR0 K = 2*(L/16), VGPR1 +1.
//     Since u[.,g] = sum_f x*Uw[g,f], B[k][n] = Uw[g0+n, k] — each lane streams
//     its own Uw row, same addressing as A.
//   C/D (16x16 f32, 8 VGPRs): VGPR r -> row r + 8*(L/16), col L%16.
// ---------------------------------------------------------------------------
__global__ __launch_bounds__(32) void u_gemm_wmma(
    const float* __restrict__ x, const float* __restrict__ Uw,
    const float* __restrict__ Ub, float* __restrict__ u) {
  const int tileR = blockIdx.x;      // over (B*N)/16 = 64
  const int tileG = blockIdx.y;      // over F2/16    = 16
  const int lane  = threadIdx.x;
  const int l16   = lane & 15;
  const int half  = lane >> 4;       // 0: K pair {0,1}; 1: K pair {2,3}

  const float* __restrict__ xrow = x  + (size_t)(tileR * 16 + l16) * kF1;
  const float* __restrict__ wrow = Uw + (size_t)(tileG * 16 + l16) * kF1;

  v8f c0 = {};
  v8f c1 = {};
  // Two interleaved accumulators to break the WMMA D->C dependency chain.
  for (int k = 0; k < kF1; k += 8) {
    const int ka = k + half * 2;
    const int kb = k + 4 + half * 2;
    v2f a0 = *(const v2f*)(xrow + ka);
    v2f b0 = *(const v2f*)(wrow + ka);
    v2f a1 = *(const v2f*)(xrow + kb);
    v2f b1 = *(const v2f*)(wrow + kb);
    c0 = __builtin_amdgcn_wmma_f32_16x16x4_f32(
        /*neg_a=*/false, a0, /*neg_b=*/false, b0,
        /*c_mod=*/(short)0, c0, /*reuse_a=*/false, /*reuse_b=*/false);
    c1 = __builtin_amdgcn_wmma_f32_16x16x4_f32(
        false, a1, false, b1, (short)0, c1, false, false);
  }

  const float bias = Ub[tileG * 16 + l16];
#pragma unroll
  for (int r = 0; r < 8; ++r) {
    const int row = r + half * 8;
    u[(size_t)(tileR * 16 + row) * kF2 + tileG * 16 + l16] = c0[r] + c1[r] + bias;
  }
}

// ---------------------------------------------------------------------------
// Kernel 2: alpha[b,n,m] = sum_f tanh(u[b,n,f] * y[b,m,f]) * Ww[f] + Wb
// Block = 256 threads (thread t <-> column m = t), each block covers
// (b, 8 consecutive n). y staged through LDS in 256x32 f-tiles, row stride
// 33 floats => bank (33*t + f) % 64 is conflict-free across a wave32.
// u rows are uniform LDS broadcasts; 8-deep accumulator amortizes each
// y/w read across 8 tanh-FMAs.
// ---------------------------------------------------------------------------
constexpr int kNChunk = 8;
constexpr int kFT     = 32;

__global__ __launch_bounds__(256) void alpha_kernel(
    const float* __restrict__ u, const float* __restrict__ y,
    const float* __restrict__ Ww, const float* __restrict__ Wb,
    float* __restrict__ alpha) {
  const int b  = blockIdx.y;
  const int n0 = blockIdx.x * kNChunk;
  const int t  = threadIdx.x;          // m index

  __shared__ float ylds[kM][kFT + 1];  // padded: conflict-free
  __shared__ float ulds[kNChunk][kF2];
  __shared__ float wlds[kF2];

  // Stage u rows for this n-chunk (coalesced: inner index == t) and Ww.
  for (int i = t; i < kNChunk * kF2; i += 256) {
    const int n = i >> 8;              // kF2 == 256
    const int f = i & 255;
    ulds[n][f] = u[(size_t)(b * kN + n0 + n) * kF2 + f];
  }
  wlds[t] = Ww[t];

  float acc[kNChunk];
#pragma unroll
  for (int n = 0; n < kNChunk; ++n) acc[n] = 0.f;

  const float* __restrict__ yb = y + (size_t)b * kM * kF2;

  for (int ft = 0; ft < kF2 / kFT; ++ft) {
    __syncthreads();
    // Cooperative load of y tile [256 m][32 f]: coalesced 32-element runs.
#pragma unroll
    for (int i = 0; i < kFT; ++i) {
      const int flat = t + i * 256;
      const int m = flat >> 5;         // / kFT
      const int f = flat & (kFT - 1);
      ylds[m][f] = yb[(size_t)m * kF2 + ft * kFT + f];
    }
    __syncthreads();

    // Prefetch next tile (global_prefetch_b8) while we crunch this one.
    if (ft + 1 < kF2 / kFT)
      __builtin_prefetch((const void*)(yb + (size_t)t * kF2 + (ft + 1) * kFT), 0, 0);

#pragma unroll 4
    for (int f = 0; f < kFT; ++f) {
      const float yv = ylds[t][f];
      const float wv = wlds[ft * kFT + f];
#pragma unroll
      for (int n = 0; n < kNChunk; ++n)
        acc[n] = __builtin_fmaf(fast_tanh(ulds[n][ft * kFT + f] * yv), wv, acc[n]);
    }
  }

  const float wb = Wb[0];
#pragma unroll
  for (int n = 0; n < kNChunk; ++n)
    alpha[(size_t)(b * kN + n0 + n) * kM + t] = acc[n] + wb;
}

// ---------------------------------------------------------------------------
// Kernel 3: maxes, softmaxes, weighted pooling. One block per batch (b).
// ---------------------------------------------------------------------------
__global__ __launch_bounds__(256) void finalize_kernel(
    const float* __restrict__ x, const float* __restrict__ y,
    const float* __restrict__ alpha, float* __restrict__ out) {
  const int b = blockIdx.x;
  const int t = threadIdx.x;

  __shared__ float rw[kN];   // row max -> xw
  __shared__ float cw[kM];   // col max -> yw
  __shared__ float red[256];

  const float* __restrict__ A = alpha + (size_t)b * kN * kM;

  // Column max over n (thread t = m): coalesced across t.
  {
    float mx = -INFINITY;
    for (int n = 0; n < kN; ++n) mx = fmaxf(mx, A[(size_t)n * kM + t]);
    cw[t] = mx;
  }
  // Row max over m (thread t = n, t < 128): each thread streams one row.
  if (t < kN) {
    float mx = -INFINITY;
    const float* __restrict__ row = A + (size_t)t * kM;
    for (int m = 0; m < kM; ++m) mx = fmaxf(mx, row[m]);
    rw[t] = mx;
  }
  __syncthreads();

  // Softmax over rw[0..127] -> xw (in place).
  red[t] = (t < kN) ? rw[t] : -INFINITY;
  __syncthreads();
  for (int s = 128; s > 0; s >>= 1) {
    if (t < s) red[t] = fmaxf(red[t], red[t + s]);
    __syncthreads();
  }
  const float gmx = red[0];
  __syncthreads();
  const float ex = (t < kN) ? __expf(rw[t] - gmx) : 0.f;
  red[t] = ex;
  __syncthreads();
  for (int s = 128; s > 0; s >>= 1) {
    if (t < s) red[t] += red[t + s];
    __syncthreads();
  }
  const float sx = red[0];
  __syncthreads();
  if (t < kN) rw[t] = ex / sx;

  // Softmax over cw[0..255] -> yw (in place).
  red[t] = cw[t];
  __syncthreads();
  for (int s = 128; s > 0; s >>= 1) {
    if (t < s) red[t] = fmaxf(red[t], red[t + s]);
    __syncthreads();
  }
  const float gmy = red[0];
  __syncthreads();
  const float ey = __expf(cw[t] - gmy);
  red[t] = ey;
  __syncthreads();
  for (int s = 128; s > 0; s >>= 1) {
    if (t < s) red[t] += red[t + s];
    __syncthreads();
  }
  const float sy = red[0];
  __syncthreads();
  cw[t] = ey / sy;
  __syncthreads();

  // Pooling (thread t = feature f; coalesced across t).
  float accx = 0.f;
  for (int n = 0; n < kN; ++n)
    accx = __builtin_fmaf(rw[n], x[(size_t)(b * kN + n) * kF1 + t], accx);
  out[(size_t)b * (kF1 + kF2) + t] = accx;

  float accy = 0.f;
  for (int m = 0; m < kM; ++m)
    accy = __builtin_fmaf(cw[m], y[(size_t)(b * kM + m) * kF2 + t], accy);
  out[(size_t)b * (kF1 + kF2) + kF1 + t] = accy;
}

// ---------------------------------------------------------------------------
extern "C" void kernel_launch(void* const* d_in, const int* in_sizes, int n_in,
                              void* d_out, int out_size, void* d_ws, size_t ws_size,
                              hipStream_t stream) {
  const float* x  = (const float*)d_in[0];
  const float* y  = (const float*)d_in[1];
  const float* Uw = (const float*)d_in[2];
  const float* Ub = (const float*)d_in[3];
  const float* Ww = (const float*)d_in[4];
  const float* Wb = (const float*)d_in[5];
  float* out = (float*)d_out;

  float* u     = (float*)d_ws;                        // [B,N,F2] = 1 MB
  float* alpha = u + (size_t)kB * kN * kF2;           // [B,N,M]  = 1 MB

  u_gemm_wmma<<<dim3((kB * kN) / 16, kF2 / 16), dim3(32), 0, stream>>>(x, Uw, Ub, u);
  alpha_kernel<<<dim3(kN / kNChunk, kB), dim3(256), 0, stream>>>(u, y, Ww, Wb, alpha);
  finalize_kernel<<<dim3(kB), dim3(256), 0, stream>>>(x, y, alpha, out);
}